// BasePointProcess_25838523253135
// MI455X (gfx1250) — compile-verified
//
#include <hip/hip_runtime.h>
#include <math.h>

typedef __attribute__((ext_vector_type(2))) float v2f;
typedef __attribute__((ext_vector_type(8))) float v8f;

#define PI_F      3.14159265358979f
#define T1_F      365.0f
#define LON0_F   (-0.42f)
#define LON1_F   (-0.30f)
#define LAT0_F    39.40f
#define LAT1_F    39.52f
#define KM_PER_LON (111.32f * 0.772f)
#define KM_PER_LAT (110.574f)
#define C_LON    (-0.36f)
#define C_LAT     39.46f
#define EPS_F     1e-5f

// ---------------------------------------------------------------------------
// Init: out[0] = -baserate = -sum(mu) * (T1-T0) * AREA * B
// (d_out is poisoned before timing; must be re-initialized every launch.)
// ---------------------------------------------------------------------------
__global__ void hawkes_init_out(const float* __restrict__ mu, float* __restrict__ out, int B) {
    if (threadIdx.x == 0 && blockIdx.x == 0) {
        float area = (LON1_F - LON0_F) * KM_PER_LON * (LAT1_F - LAT0_F) * KM_PER_LAT;
        float msum = mu[0] + mu[1] + mu[2] + mu[3];
        out[0] = -msum * T1_F * area * (float)B;
    }
}

// ---------------------------------------------------------------------------
// Main: one wave32 per (batch, 16-row tile). Causal 16x16 tiles of the
// pairwise exponent are computed with V_WMMA_F32_16X16X4_F32 (rank-4 form).
// ---------------------------------------------------------------------------
__global__ __launch_bounds__(256)
void hawkes_wmma_kernel(const float4* __restrict__ X,   // [B, L] of (t, cls, lon, lat)
                        const float*  __restrict__ mu,  // [4]
                        const float*  __restrict__ alpha_p,
                        const float*  __restrict__ beta_p,
                        const float*  __restrict__ sigma_p,
                        float* __restrict__ out,
                        int L, int ntiles)               // ntiles = L/16 per batch
{
    const float alpha  = alpha_p[0];
    const float beta   = beta_p[0];
    const float sigma  = sigma_p[0];
    const float inv2s2 = 1.0f / (2.0f * sigma * sigma);   // 1/(2σ²)
    const float invs2  = 2.0f * inv2s2;                   // 1/σ²
    const float scale  = alpha * beta * inv2s2 / PI_F;    // αβ / (π·2σ²)

    const int wave  = (int)((blockIdx.x * blockDim.x + threadIdx.x) >> 5);
    const int lane  = (int)(threadIdx.x & 31u);
    const int half  = lane >> 4;     // 0: lanes 0-15, 1: lanes 16-31
    const int l16   = lane & 15;

    const int batch = wave / ntiles;
    const int I     = wave % ntiles;
    const int i0    = I * 16;
    const float4* Xb = X + (size_t)batch * (size_t)L;

    // ---- A fragment: rows i0..i0+15, 16x4 f32 layout (2 VGPRs) ----
    // lanes 0-15 : a.x = A[m][K=0], a.y = A[m][K=1]   (m = lane)
    // lanes 16-31: a.x = A[m][K=2], a.y = A[m][K=3]   (m = lane-16)
    const float4 xi  = Xb[i0 + l16];
    const float  sxi = xi.z - C_LON;
    const float  syi = xi.w - C_LAT;
    const float  normi = (sxi * sxi + syi * syi) * inv2s2;
    v2f a;
    a.x = half ? (sxi * invs2) : 1.0f;
    a.y = half ? (syi * invs2) : (-beta * xi.x - normi);

    float acc[8];
#pragma unroll
    for (int k = 0; k < 8; ++k) acc[k] = 0.0f;

    for (int J = 0; J <= I; ++J) {
        const int j0 = J * 16;
        // Prefetch next column tile (global_prefetch_b8 path)
        if (j0 + 16 < L) __builtin_prefetch(&Xb[j0 + 16 + l16], 0, 1);

        // ---- B fragment: cols j0..j0+15, 4x16 f32 layout (2 VGPRs) ----
        const float4 xj  = Xb[j0 + l16];
        const float  sxj = xj.z - C_LON;
        const float  syj = xj.w - C_LAT;
        const float  normj = (sxj * sxj + syj * syj) * inv2s2;
        v2f bf;
        bf.x = half ? sxj : (beta * xj.x - normj);
        bf.y = half ? syj : 1.0f;

        // Full-EXEC WMMA: E-tile = A x B, C = 0
        v8f c = {};
        c = __builtin_amdgcn_wmma_f32_16x16x4_f32(
                /*neg_a=*/false, a, /*neg_b=*/false, bf,
                /*c_mod=*/(short)0, c, /*reuse_a=*/false, /*reuse_b=*/false);

        // C/D layout: this lane holds column n = l16; element k is row
        // m = k + 8*half.  Mask: valid(t_j>0) and causal (j < i).
        const float vj   = (xj.x > 0.0f) ? 1.0f : 0.0f;
        const bool  offd = (J < I);
#pragma unroll
        for (int k = 0; k < 8; ++k) {
            const int  m    = k + (half << 3);
            const bool keep = offd | (l16 < m);       // strict lower triangle on diagonal tile
            const float e   = __expf(c[k]);
            acc[k] += keep ? (e * vj) : 0.0f;
        }
    }

    // ---- Row reduction across the 16 lanes of each half (xor 1,2,4,8) ----
#pragma unroll
    for (int k = 0; k < 8; ++k) {
        float v = acc[k];
#pragma unroll
        for (int off = 1; off < 16; off <<= 1)
            v += __shfl_xor(v, off, 32);
        acc[k] = v;   // every lane in the half now has rowsum of row (k + 8*half)
    }

    const float mu0 = mu[0], mu1 = mu[1], mu2 = mu[2], mu3 = mu[3];

    // ---- Per-row log-intensity; row r data lives in lanes r and 16+r ----
    float total = 0.0f;
#pragma unroll
    for (int r = 0; r < 16; ++r) {
        const float rowsum = __shfl(acc[r & 7], (r < 8) ? 0 : 16, 32);
        const float t_r    = __shfl(xi.x, r, 32);
        const float cls_r  = __shfl(xi.y, r, 32);
        const int   ci     = (int)cls_r;
        const float m      = (ci == 0) ? mu0 : (ci == 1) ? mu1 : (ci == 2) ? mu2 : mu3;
        const float lam    = scale * rowsum + m;
        total += (t_r > 0.0f) ? __logf(lam + EPS_F) : 0.0f;
    }

    if (lane == 0) atomicAdd(out, total);
}

// ---------------------------------------------------------------------------
extern "C" void kernel_launch(void* const* d_in, const int* in_sizes, int n_in,
                              void* d_out, int out_size, void* d_ws, size_t ws_size,
                              hipStream_t stream) {
    const float* X     = (const float*)d_in[0];   // [B, L, 4] f32
    const float* mu    = (const float*)d_in[1];   // [4] f32
    const float* alpha = (const float*)d_in[2];   // scalar
    const float* beta  = (const float*)d_in[3];   // scalar
    const float* sigma = (const float*)d_in[4];   // scalar
    float* out = (float*)d_out;

    const int B  = 4;                      // fixed by the reference setup
    const int BL = in_sizes[0] / 4;        // B*L
    const int L  = BL / B;                 // 2048
    const int ntiles = L / 16;             // 128 row-tiles per batch

    hawkes_init_out<<<1, 32, 0, stream>>>(mu, out, B);

    const int totalWaves = B * ntiles;     // 512 wave32's
    const int threads = 256;               // 8 waves per block
    const int blocks  = (totalWaves * 32 + threads - 1) / threads;
    hawkes_wmma_kernel<<<blocks, threads, 0, stream>>>(
        (const float4*)X, mu, alpha, beta, sigma, out, L, ntiles);
}